// SuperPointEncoder_8409545966068
// MI455X (gfx1250) — compile-verified
//
#include <hip/hip_runtime.h>

// ---- problem constants (from reference) ----
#define BATCH 4
#define NPTS 16384
#define NOUT 1170          // NPTS / FPS_FACTOR(14)
#define DIN 128
#define DOUT 256

// ---- FPS kernel config ----
#define FPS_THREADS 1024
#define NPT (NPTS / FPS_THREADS)   // 16 points per thread, kept in registers

// ---- GEMM kernel config ----
#define RT ((NOUT + 15) / 16)          // 74 row tiles per batch
#define CT (DOUT / 16)                 // 16 col tiles
#define TILES_PER_B (RT * CT)          // 1184
#define TOT_TILES (BATCH * TILES_PER_B)// 4736
#define GEMM_WPB 16                    // waves per block
#define GEMM_BLOCK (GEMM_WPB * 32)     // 512 threads
#define GEMM_GRID (TOT_TILES / GEMM_WPB) // 296 blocks (exact)

typedef __attribute__((ext_vector_type(2))) float v2f;
typedef __attribute__((ext_vector_type(8))) float v8f;

// ============================================================================
// Kernel 1: farthest point sampling. One workgroup per batch; points live in
// registers; per-step argmax via wave32 shuffle reduce + one LDS stage.
// Emits sel_idx (ws), query_c and query_v outputs as selections happen.
// ============================================================================
__global__ __launch_bounds__(FPS_THREADS)
void fps_kernel(const float* __restrict__ pcd_x,
                const float* __restrict__ pcd_v,
                int*   __restrict__ sel_idx,   // [B, NOUT]
                float* __restrict__ qc,        // [B, NOUT, 3]
                float* __restrict__ qv)        // [B, NOUT, 3, 3]
{
    const int b    = blockIdx.x;
    const int tid  = threadIdx.x;
    const int lane = tid & 31;
    const int wid  = tid >> 5;

    __shared__ float wbd[32], wbx[32], wby[32], wbz[32];
    __shared__ int   wbi[32];
    __shared__ float curX, curY, curZ;

    // own 16 points in registers
    float px[NPT], py[NPT], pz[NPT], dist[NPT];
    const float* base = pcd_x + (size_t)b * NPTS * 3;
#pragma unroll
    for (int t = 0; t < NPT; ++t) {
        const int j = tid + t * FPS_THREADS;
        px[t]  = base[j * 3 + 0];
        py[t]  = base[j * 3 + 1];
        pz[t]  = base[j * 3 + 2];
        dist[t] = 1e10f;
    }

    // step 0: deterministic start at index 0
    if (tid == 0) {
        curX = base[0]; curY = base[1]; curZ = base[2];
        sel_idx[b * NOUT + 0] = 0;
        float* qcp = qc + ((size_t)b * NOUT + 0) * 3;
        qcp[0] = base[0]; qcp[1] = base[1]; qcp[2] = base[2];
        const float* vs = pcd_v + (size_t)b * NPTS * 9;
        float* vd = qv + ((size_t)b * NOUT + 0) * 9;
#pragma unroll
        for (int q = 0; q < 9; ++q) vd[q] = vs[q];
    }
    __syncthreads();

    for (int s = 1; s < NOUT; ++s) {
        const float lx = curX, ly = curY, lz = curZ;
        float bd = -1.0f, bx = 0.f, by = 0.f, bz = 0.f;
        int bi = 0;
#pragma unroll
        for (int t = 0; t < NPT; ++t) {
            const float dx = px[t] - lx, dy = py[t] - ly, dz = pz[t] - lz;
            float d = dx * dx + dy * dy + dz * dz;
            d = fminf(d, dist[t]);
            dist[t] = d;
            if (d > bd) { bd = d; bi = tid + t * FPS_THREADS; bx = px[t]; by = py[t]; bz = pz[t]; }
        }
        // wave32 argmax reduce (tie -> lowest index, matching jnp.argmax)
#pragma unroll
        for (int off = 16; off > 0; off >>= 1) {
            const float od = __shfl_xor(bd, off, 32);
            const int   oi = __shfl_xor(bi, off, 32);
            const float ox = __shfl_xor(bx, off, 32);
            const float oy = __shfl_xor(by, off, 32);
            const float oz = __shfl_xor(bz, off, 32);
            if (od > bd || (od == bd && oi < bi)) { bd = od; bi = oi; bx = ox; by = oy; bz = oz; }
        }
        if (lane == 0) { wbd[wid] = bd; wbi[wid] = bi; wbx[wid] = bx; wby[wid] = by; wbz[wid] = bz; }
        __syncthreads();
        if (wid == 0) {
            bd = wbd[lane]; bi = wbi[lane]; bx = wbx[lane]; by = wby[lane]; bz = wbz[lane];
#pragma unroll
            for (int off = 16; off > 0; off >>= 1) {
                const float od = __shfl_xor(bd, off, 32);
                const int   oi = __shfl_xor(bi, off, 32);
                const float ox = __shfl_xor(bx, off, 32);
                const float oy = __shfl_xor(by, off, 32);
                const float oz = __shfl_xor(bz, off, 32);
                if (od > bd || (od == bd && oi < bi)) { bd = od; bi = oi; bx = ox; by = oy; bz = oz; }
            }
            if (lane == 0) {
                curX = bx; curY = by; curZ = bz;
                sel_idx[b * NOUT + s] = bi;
                float* qcp = qc + ((size_t)b * NOUT + s) * 3;
                qcp[0] = bx; qcp[1] = by; qcp[2] = bz;
                const float* vs = pcd_v + ((size_t)b * NPTS + bi) * 9;
                float* vd = qv + ((size_t)b * NOUT + s) * 9;
#pragma unroll
                for (int q = 0; q < 9; ++q) vd[q] = vs[q];
            }
        }
        __syncthreads();
    }
}

// ============================================================================
// Kernel 2: gather selected feature rows and project with W (fp32 WMMA).
// One 16x16 output tile per wave; K=128 -> 32 chained v_wmma_f32_16x16x4_f32.
// ============================================================================
__global__ __launch_bounds__(GEMM_BLOCK)
void gather_gemm_kernel(const float* __restrict__ tgt,
                        const float* __restrict__ W,     // [DIN, DOUT] row-major
                        const float* __restrict__ bias,  // [DOUT]
                        const int*   __restrict__ sel_idx,
                        float* __restrict__ qf)          // [B, NOUT, DOUT]
{
    const int lane = threadIdx.x & 31;
    const int wave = blockIdx.x * GEMM_WPB + (threadIdx.x >> 5);
    const int b  = wave / TILES_PER_B;
    const int r  = wave % TILES_PER_B;
    const int tm = r / CT;
    const int tn = r % CT;
    const int half = lane >> 4;  // which half-wave
    const int l16  = lane & 15;

    // A fragment: row owned by this lane (clamped for the partial last tile)
    const int  row    = tm * 16 + l16;
    const int  rclamp = row < NOUT ? row : NOUT - 1;
    const int  gidx   = sel_idx[b * NOUT + rclamp];
    const float* arow = tgt + ((size_t)b * NPTS + gidx) * DIN;
    const int  n      = tn * 16 + l16;

    v8f c = {};
#pragma unroll
    for (int k0 = 0; k0 < DIN; k0 += 4) {
        const int k = k0 + 2 * half;
        // A 16x4 layout: lanes 0-15 -> K = k0,k0+1 ; lanes 16-31 -> K = k0+2,k0+3
        v2f a;
        a.x = arow[k];
        a.y = arow[k + 1];
        // B 4x16 layout (N striped over lanes), same K split per half-wave
        v2f bb;
        bb.x = W[(size_t)k * DOUT + n];
        bb.y = W[(size_t)(k + 1) * DOUT + n];
        c = __builtin_amdgcn_wmma_f32_16x16x4_f32(
                /*neg_a=*/false, a, /*neg_b=*/false, bb,
                /*c_mod=*/(short)0, c, /*reuse_a=*/false, /*reuse_b=*/false);
    }

    const float bv = bias[n];
#pragma unroll
    for (int rr = 0; rr < 8; ++rr) {
        const int m    = rr + 8 * half;            // C/D layout: VGPR rr -> M
        const int orow = tm * 16 + m;
        if (orow < NOUT) {
            qf[((size_t)b * NOUT + orow) * DOUT + n] = c[rr] + bv;
        }
    }
}

// ============================================================================
extern "C" void kernel_launch(void* const* d_in, const int* in_sizes, int n_in,
                              void* d_out, int out_size, void* d_ws, size_t ws_size,
                              hipStream_t stream) {
    const float* tgt   = (const float*)d_in[0];  // [B, N, DIN]
    const float* pcd_x = (const float*)d_in[1];  // [B, N, 3]
    const float* pcd_v = (const float*)d_in[2];  // [B, N, 3, 3]
    const float* W     = (const float*)d_in[3];  // [DIN, DOUT]
    const float* bias  = (const float*)d_in[4];  // [DOUT]

    float* out = (float*)d_out;
    float* qf = out;                                    // [B, NOUT, DOUT]
    float* qc = qf + (size_t)BATCH * NOUT * DOUT;       // [B, NOUT, 3]
    float* qv = qc + (size_t)BATCH * NOUT * 3;          // [B, NOUT, 3, 3]
    int*   sel = (int*)d_ws;                            // [B, NOUT]

    fps_kernel<<<BATCH, FPS_THREADS, 0, stream>>>(pcd_x, pcd_v, sel, qc, qv);
    gather_gemm_kernel<<<GEMM_GRID, GEMM_BLOCK, 0, stream>>>(tgt, W, bias, sel, qf);
}